// STTC_L0_NN_17686675324931
// MI455X (gfx1250) — compile-verified
//
#include <hip/hip_runtime.h>
#include <stdint.h>

// ---------------------------------------------------------------------------
// STTC tensor-ring completion sweep for MI455X (gfx1250, wave32).
// f32 WMMA (v_wmma_f32_16x16x4_f32) for all GEMMs; async gather-to-LDS
// staging for the split-K GEMM; register-resident B for the reconstruction.
// ---------------------------------------------------------------------------

typedef float v2f __attribute__((ext_vector_type(2)));
typedef float v8f __attribute__((ext_vector_type(8)));

#define I0 288
#define I1 512
#define I2 256
#define S0 131072   // I1*I2 (stride of dim0)
#define TOT 37748736L

__device__ __forceinline__ v8f wmma4(v2f a, v2f b, v8f c) {
  return __builtin_amdgcn_wmma_f32_16x16x4_f32(false, a, false, b, (short)0, c,
                                               false, false);
}

__device__ __forceinline__ void async_gather_b32(unsigned lds_addr,
                                                 const float* gptr) {
  unsigned long long ga = (unsigned long long)(uintptr_t)gptr;
  asm volatile("global_load_async_to_lds_b32 %0, %1, off"
               :: "v"(lds_addr), "v"(ga) : "memory");
}

// ---- subchain, row-major:  C[jk*64 + a*8+b] = sum_c Ga[b,j,c]*Gb[c,k,a] ----
__global__ void k_build_c_rm(const float* __restrict__ Ga,
                             const float* __restrict__ Gb,
                             int Ia, int Ib, float* __restrict__ C, int total) {
  int t = blockIdx.x * 256 + threadIdx.x;
  if (t >= total) return;
  int col = t & 63;
  int jk  = t >> 6;
  int a = col >> 3, b = col & 7;
  int j = jk / Ib;
  int k = jk - j * Ib;
  const float* ga = Ga + (b * Ia + j) * 8;
  float s = 0.f;
#pragma unroll
  for (int c = 0; c < 8; ++c) s += ga[c] * Gb[(c * Ib + k) * 8 + a];
  C[t] = s;
}

// ---- subchain, transposed: Ct[col*J + jk] (contiguous in k for WMMA B) -----
__global__ void k_build_ct(const float* __restrict__ Ga,
                           const float* __restrict__ Gb,
                           int Ia, int Ib, float* __restrict__ Ct, int J) {
  int jk  = blockIdx.x * 256 + threadIdx.x;
  int col = blockIdx.y;                       // 0..63
  int a = col >> 3, b = col & 7;
  int j = jk / Ib;
  int k = jk - j * Ib;
  const float* ga = Ga + (b * Ia + j) * 8;
  float s = 0.f;
#pragma unroll
  for (int c = 0; c < 8; ++c) s += ga[c] * Gb[(c * Ib + k) * 8 + a];
  Ct[(long)col * J + jk] = s;
}

__global__ void k_zero(float* __restrict__ p, int n) {
  int t = blockIdx.x * 256 + threadIdx.x;
  if (t < n) p[t] = 0.f;
}

// ---- NC = N_m @ C  (split-K; N tile async-staged in LDS, shared by 4 waves,
//      two 16-row m-tiles per block; Ct is [64][J] so B loads are b64) -------
template <int MODE>
__global__ void k_nc(const float* __restrict__ Xrec,
                     const float* __restrict__ Aten,
                     const float* __restrict__ Ct,
                     float* __restrict__ NC, int J, int kc) {
  __shared__ float xt[32 * 128];
  __shared__ float at[32 * 128];
  const int lane = threadIdx.x;
  const int tid  = threadIdx.y * 32 + lane;   // 0..127
  const int half = lane >> 4;
  const int lo   = lane & 15;
  const int tn   = threadIdx.y;               // 0..3  (16 output cols each)
  const int mbase = blockIdx.y * 32;
  const int kbase = blockIdx.x * kc;
  const float* ctrow = Ct + (long)(tn * 16 + lo) * J;

  v8f acc0 = {0.f, 0.f, 0.f, 0.f, 0.f, 0.f, 0.f, 0.f};
  v8f acc1 = {0.f, 0.f, 0.f, 0.f, 0.f, 0.f, 0.f, 0.f};

  for (int ks = 0; ks < kc; ks += 128) {
    // stage raw Xrec / A 32x128 tiles via per-lane async gather to LDS
    for (int e = tid; e < 32 * 128; e += 128) {
      int row = e >> 7, kk = e & 127;
      int m = mbase + row;
      int k = kbase + ks + kk;
      int idx;
      if (MODE == 0) {
        idx = m * S0 + k;
      } else if (MODE == 1) {
        int n2 = k / I0, n0 = k - n2 * I0;
        idx = n0 * S0 + m * I2 + n2;
      } else {
        int n0 = k / I1, n1 = k - n0 * I1;
        idx = n0 * S0 + n1 * I2 + m;
      }
      async_gather_b32((unsigned)(uintptr_t)&xt[e], Xrec + idx);
      async_gather_b32((unsigned)(uintptr_t)&at[e], Aten + idx);
    }
    asm volatile("s_wait_asynccnt 0x0" ::: "memory");
    __syncthreads();
    for (int kk = 0; kk < 128; kk += 4) {
      int k = kbase + ks + kk;
      v2f b   = *(const v2f*)(ctrow + k + 2 * half);
      v2f x0  = *(const v2f*)(&xt[lo * 128 + kk + 2 * half]);
      v2f a0  = *(const v2f*)(&at[lo * 128 + kk + 2 * half]);
      v2f x1  = *(const v2f*)(&xt[(16 + lo) * 128 + kk + 2 * half]);
      v2f a1  = *(const v2f*)(&at[(16 + lo) * 128 + kk + 2 * half]);
      acc0 = wmma4(x0 - a0, b, acc0);
      acc1 = wmma4(x1 - a1, b, acc1);
    }
    __syncthreads();
  }
#pragma unroll
  for (int p = 0; p < 8; ++p) {
    int c = tn * 16 + lo;
    atomicAdd(&NC[(mbase + p + 8 * half) * 64 + c], acc0[p]);
    atomicAdd(&NC[(mbase + 16 + p + 8 * half) * 64 + c], acc1[p]);
  }
}

// ---- CtC = C^T @ C  (split-K; both operands contiguous from Ct) ------------
__global__ void k_ctc(const float* __restrict__ Ct, float* __restrict__ CtC,
                      int J, int kc) {
  const int lane = threadIdx.x;
  const int half = lane >> 4;
  const int lo   = lane & 15;
  const int tn   = threadIdx.y;   // 0..3
  const int tm   = blockIdx.y;    // 0..3
  const int kbase = blockIdx.x * kc;
  const float* arow = Ct + (long)(tm * 16 + lo) * J;
  const float* brow = Ct + (long)(tn * 16 + lo) * J;

  v8f acc = {0.f, 0.f, 0.f, 0.f, 0.f, 0.f, 0.f, 0.f};
  for (int kk = 0; kk < kc; kk += 4) {
    int k = kbase + kk + 2 * half;
    v2f a = *(const v2f*)(arow + k);
    v2f b = *(const v2f*)(brow + k);
    acc = wmma4(a, b, acc);
  }
#pragma unroll
  for (int p = 0; p < 8; ++p)
    atomicAdd(&CtC[(tm * 16 + p + 8 * half) * 64 + tn * 16 + lo], acc[p]);
}

// ---- Binv = inv(CtC + (alpha + 0.5 mu + 0.5 ni) I), Gauss-Jordan in LDS ----
__global__ void k_inv64(const float* __restrict__ CtC, float* __restrict__ Binv,
                        const float* __restrict__ alpha,
                        const float* __restrict__ ni,
                        const float* __restrict__ mu, int mode) {
  __shared__ float M[64 * 128];
  __shared__ float fcol[64];
  const int tid = threadIdx.x;  // 256 threads
  const float lam = alpha[mode] + 0.5f * (mu[mode] + ni[mode]);
  for (int e = tid; e < 64 * 128; e += 256) {
    int r = e >> 7, c = e & 127;
    float v;
    if (c < 64) v = CtC[r * 64 + c] + ((r == c) ? lam : 0.f);
    else        v = ((c - 64) == r) ? 1.f : 0.f;
    M[e] = v;
  }
  __syncthreads();
  for (int k = 0; k < 64; ++k) {
    if (tid < 64) fcol[tid] = M[tid * 128 + k];
    __syncthreads();
    float inv = 1.0f / fcol[k];
    if (tid < 128) M[k * 128 + tid] *= inv;
    __syncthreads();
    for (int e = tid; e < 64 * 128; e += 256) {
      int r = e >> 7, c = e & 127;
      if (r != k) M[e] -= fcol[r] * M[k * 128 + c];
    }
    __syncthreads();
  }
  for (int e = tid; e < 4096; e += 256) {
    int r = e >> 6, c = e & 63;
    Binv[e] = M[r * 128 + 64 + c];
  }
}

// ---- WMMA GEMM  D(M x 64) = A(M x K) @ BT^T, BT given as (64 x K) row-major
//      (for Binv we pass Binv itself: SPD inverse is symmetric) --------------
__global__ void k_gemm_n64t(const float* __restrict__ A, int lda,
                            const float* __restrict__ BT, int ldbt,
                            float* __restrict__ D, int K) {
  const int lane = threadIdx.x;
  const int half = lane >> 4;
  const int lo   = lane & 15;
  const int tn   = threadIdx.y;   // 0..3
  const int tm   = blockIdx.y;
  const float* arow = A + (tm * 16 + lo) * lda + 2 * half;
  const float* brow = BT + (tn * 16 + lo) * ldbt + 2 * half;
  v8f acc = {0.f, 0.f, 0.f, 0.f, 0.f, 0.f, 0.f, 0.f};
  for (int k0 = 0; k0 < K; k0 += 4) {
    v2f a = *(const v2f*)(arow + k0);
    v2f b = *(const v2f*)(brow + k0);
    acc = wmma4(a, b, acc);
  }
#pragma unroll
  for (int p = 0; p < 8; ++p)
    D[(tm * 16 + p + 8 * half) * 64 + tn * 16 + lo] = acc[p];
}

// ---- small elementwise kernels ---------------------------------------------
__global__ void k_init(const float* __restrict__ Gin, float* __restrict__ G,
                       float* __restrict__ Z, float* __restrict__ Q,
                       float* __restrict__ Y, float* __restrict__ U, int n) {
  int t = blockIdx.x * 256 + threadIdx.x;
  if (t >= n) return;
  float g = Gin[t];
  G[t] = g; Z[t] = g; Q[t] = g; Y[t] = 0.f; U[t] = 0.f;
}

__global__ void k_amat(const float* __restrict__ NC, const float* __restrict__ Z,
                       const float* __restrict__ Y, const float* __restrict__ Q,
                       const float* __restrict__ U, const float* __restrict__ G,
                       float* __restrict__ Amat,
                       const float* __restrict__ alpha,
                       const float* __restrict__ ni,
                       const float* __restrict__ mu, int mode, int n) {
  int t = blockIdx.x * 256 + threadIdx.x;
  if (t >= n) return;
  Amat[t] = NC[t] + 0.5f * mu[mode] * Z[t] - 0.5f * Y[t]
          + 0.5f * ni[mode] * Q[t] - 0.5f * U[t] + alpha[mode] * G[t];
}

__global__ void k_aq(const float* __restrict__ W, const float* __restrict__ Bp,
                     float* __restrict__ Rm, int n) {
  int t = blockIdx.x * 256 + threadIdx.x;
  if (t < n) Rm[t] = W[t] * Rm[t] + Bp[t];
}

// V^T (64 x In): VT[col*In + row] = ni*G[row,col] + U[row,col]
__global__ void k_vt(const float* __restrict__ G, const float* __restrict__ U,
                     float* __restrict__ VT, const float* __restrict__ ni,
                     int mode, int In, int n) {
  int t = blockIdx.x * 256 + threadIdx.x;
  if (t >= n) return;
  int row = t >> 6, col = t & 63;
  VT[col * In + row] = ni[mode] * G[t] + U[t];
}

__global__ void k_post(const float* __restrict__ G, const float* __restrict__ Q,
                       float* __restrict__ Z, float* __restrict__ U,
                       float* __restrict__ Y, const float* __restrict__ mu,
                       const float* __restrict__ ni, int mode, int n) {
  int t = blockIdx.x * 256 + threadIdx.x;
  if (t >= n) return;
  float g = G[t];
  float m = mu[mode], nn = ni[mode];
  float z = g + Y[t] / m;            // uses OLD Y (torch order)
  z = z > 0.f ? z : 0.f;
  U[t] += nn * (g - Q[t]);           // new Q, old U
  Y[t] += m * (g - z);               // old Y, new Z
  Z[t] = z;
}

__global__ void k_gtr(const float* __restrict__ G, float* __restrict__ Gtr,
                      int In, int n) {
  int t = blockIdx.x * 256 + threadIdx.x;
  if (t >= n) return;
  int col = t & 63, nrow = t >> 6;
  int a = col >> 3, b = col & 7;
  Gtr[(a * In + nrow) * 8 + b] = G[t];
}

__global__ void k_anew(const float* __restrict__ Xrec,
                       const float* __restrict__ Gcat,
                       const float* __restrict__ Om,
                       const float* __restrict__ scale,
                       float* __restrict__ Aout, long n) {
  long t = (long)blockIdx.x * 256 + threadIdx.x;
  if (t >= n) return;
  float ro = (Xrec[t] - Gcat[t]) * Om[t];
  Aout[t] = (fabsf(ro) > scale[0]) ? ro : 0.f;
}

__global__ void k_scale(const float* __restrict__ s, float* __restrict__ o) {
  o[0] = s[0] * 0.9f;
}

// ---- X_rec_new = (G_m0 @ C0^T)*(1-Om) + (X*Om)*Om --------------------------
// B fragment (16 cols x 64 k) held in registers; all 18 m-tiles looped per
// wave so C0 is streamed from HBM exactly once.
__global__ void k_recon(const float* __restrict__ G0,
                        const float* __restrict__ C0,
                        const float* __restrict__ X,
                        const float* __restrict__ Om,
                        float* __restrict__ Xout) {
  const int lane = threadIdx.x;
  const int half = lane >> 4;
  const int lo   = lane & 15;
  const int tn   = blockIdx.x * 4 + threadIdx.y;  // 0..8191 (J0/16 tiles)
  v2f bf[16];
  const float* brow = C0 + (long)(tn * 16 + lo) * 64 + 2 * half;
#pragma unroll
  for (int s = 0; s < 16; ++s) bf[s] = *(const v2f*)(brow + s * 4);
  for (int tm = 0; tm < I0 / 16; ++tm) {
    v8f acc = {0.f, 0.f, 0.f, 0.f, 0.f, 0.f, 0.f, 0.f};
    const float* arow = G0 + (tm * 16 + lo) * 64 + 2 * half;
#pragma unroll
    for (int s = 0; s < 16; ++s) {
      v2f a = *(const v2f*)(arow + s * 4);
      acc = wmma4(a, bf[s], acc);
    }
#pragma unroll
    for (int p = 0; p < 8; ++p) {
      int gm = tm * 16 + p + 8 * half;
      long idx = (long)gm * S0 + tn * 16 + lo;
      float om = Om[idx];
      Xout[idx] = acc[p] * (1.f - om) + X[idx] * om * om;
    }
  }
}

// ---------------------------------------------------------------------------
extern "C" void kernel_launch(void* const* d_in, const int* in_sizes, int n_in,
                              void* d_out, int out_size, void* d_ws,
                              size_t ws_size, hipStream_t stream) {
  (void)in_sizes; (void)n_in; (void)out_size; (void)ws_size;
  const int I[3] = {I0, I1, I2};

  const float* X     = (const float*)d_in[0];
  const float* Om    = (const float*)d_in[1];
  const float* Xrec  = (const float*)d_in[2];
  const float* Aten  = (const float*)d_in[3];
  const float* Gcat  = (const float*)d_in[4];
  const float* scale = (const float*)d_in[5];
  const float* alpha = (const float*)d_in[21];
  const float* ni    = (const float*)d_in[22];
  const float* mu    = (const float*)d_in[23];

  float* out = (float*)d_out;
  long off = 0;
  float* o_Xrec = out;          off += TOT;
  float* o_Anew = out + off;    off += TOT;
  float* o_scl  = out + off;    off += 1;
  float* o_Gm[3];  for (int i = 0; i < 3; ++i) { o_Gm[i]  = out + off; off += 64L * I[i]; }
  float* o_Gtr[3]; for (int i = 0; i < 3; ++i) { o_Gtr[i] = out + off; off += 64L * I[i]; }
  float* o_Rm[3];  for (int i = 0; i < 3; ++i) { o_Rm[i]  = out + off; off += (long)I[i] * I[i]; }

  float* w = (float*)d_ws;
  float* wC    = w; w += 147456L * 64;   // subchain matrix (max J), either layout
  float* wNC   = w; w += 512 * 64;
  float* wCtC  = w; w += 4096;
  float* wBinv = w; w += 4096;
  float* wAmat = w; w += 512 * 64;
  float* wZ    = w; w += 512 * 64;
  float* wQ    = w; w += 512 * 64;
  float* wY    = w; w += 512 * 64;
  float* wU    = w; w += 512 * 64;
  float* wVT   = w; w += 512 * 64;
  float* wGtr[3]; for (int i = 0; i < 3; ++i) { wGtr[i] = w; w += 64L * I[i]; }

  dim3 wv(32, 4);

  // working copies of the TR cores (updated mode-by-mode)
  for (int j = 0; j < 3; ++j)
    hipMemcpyAsync(wGtr[j], d_in[9 + j], 64L * I[j] * sizeof(float),
                   hipMemcpyDeviceToDevice, stream);

  for (int i = 0; i < 3; ++i) {
    const int Ia = I[(i + 1) % 3], Ib = I[(i + 2) % 3];
    const int J = Ia * Ib;
    const int n64 = I[i] * 64;

    // transposed subchain Ct[64][J]
    k_build_ct<<<dim3(J / 256, 64), 256, 0, stream>>>(wGtr[(i + 1) % 3],
                                                      wGtr[(i + 2) % 3],
                                                      Ia, Ib, wC, J);
    k_zero<<<n64 / 256, 256, 0, stream>>>(wNC, n64);
    k_zero<<<16, 256, 0, stream>>>(wCtC, 4096);

    dim3 gnc(J / 1024, I[i] / 32);
    if (i == 0)      k_nc<0><<<gnc, wv, 0, stream>>>(Xrec, Aten, wC, wNC, J, 1024);
    else if (i == 1) k_nc<1><<<gnc, wv, 0, stream>>>(Xrec, Aten, wC, wNC, J, 1024);
    else             k_nc<2><<<gnc, wv, 0, stream>>>(Xrec, Aten, wC, wNC, J, 1024);

    k_ctc<<<dim3(J / 2048, 4), wv, 0, stream>>>(wC, wCtC, J, 2048);
    k_inv64<<<1, 256, 0, stream>>>(wCtC, wBinv, alpha, ni, mu, i);

    k_init<<<n64 / 256, 256, 0, stream>>>((const float*)d_in[6 + i], o_Gm[i],
                                          wZ, wQ, wY, wU, n64);
    hipMemcpyAsync(o_Rm[i], d_in[12 + i], (long)I[i] * I[i] * sizeof(float),
                   hipMemcpyDeviceToDevice, stream);

    for (int t = 0; t < 3; ++t) {
      k_amat<<<n64 / 256, 256, 0, stream>>>(wNC, wZ, wY, wQ, wU, o_Gm[i], wAmat,
                                            alpha, ni, mu, i, n64);
      // G_new = Amat @ Binv  (Binv symmetric -> usable as BT directly)
      k_gemm_n64t<<<dim3(1, I[i] / 16), wv, 0, stream>>>(wAmat, 64, wBinv, 64,
                                                         o_Gm[i], 64);
      // Aq = W .* R_pre + B  (in place -> becomes new R_m)
      k_aq<<<(I[i] * I[i]) / 256, 256, 0, stream>>>((const float*)d_in[15 + i],
                                                    (const float*)d_in[18 + i],
                                                    o_Rm[i], I[i] * I[i]);
      k_vt<<<n64 / 256, 256, 0, stream>>>(o_Gm[i], wU, wVT, ni, i, I[i], n64);
      // Q = Aq @ V  with V^T supplied
      k_gemm_n64t<<<dim3(1, I[i] / 16), wv, 0, stream>>>(o_Rm[i], I[i], wVT,
                                                         I[i], wQ, I[i]);
      k_post<<<n64 / 256, 256, 0, stream>>>(o_Gm[i], wQ, wZ, wU, wY, mu, ni, i,
                                            n64);
    }
    k_gtr<<<n64 / 256, 256, 0, stream>>>(o_Gm[i], wGtr[i], I[i], n64);
    hipMemcpyAsync(o_Gtr[i], wGtr[i], (size_t)n64 * sizeof(float),
                   hipMemcpyDeviceToDevice, stream);
  }

  // A_new / scale_new use the OLD G_cat and scale (inputs)
  k_anew<<<TOT / 256, 256, 0, stream>>>(Xrec, Gcat, Om, scale, o_Anew, TOT);
  k_scale<<<1, 1, 0, stream>>>(scale, o_scl);

  // reconstruction with UPDATED cores: row-major mode-0 subchain, fused blend
  const int J0 = I1 * I2;  // 131072
  k_build_c_rm<<<(J0 * 64) / 256, 256, 0, stream>>>(wGtr[1], wGtr[2], I1, I2,
                                                    wC, J0 * 64);
  k_recon<<<dim3(J0 / 64, 1), wv, 0, stream>>>(o_Gm[0], wC, X, Om, o_Xrec);
}